// NNModel_17617955848270
// MI455X (gfx1250) — compile-verified
//
#include <hip/hip_runtime.h>
#include <stdint.h>

#define B_   4
#define S_   512
#define H_   768
#define C_   100
#define K_   100
#define LDC  112          // padded xl leading dim (7 * 16)
#define NEGV (-1e9f)

typedef float v2f __attribute__((ext_vector_type(2)));
typedef float v8f __attribute__((ext_vector_type(8)));

// ws layout (bytes):
//  [0,16)    uint32 thresh_bits[B]   (atomicMin of float bit patterns)
//  [16,32)   float  wsum[B]
//  [32,48)   uint32 wcnt[B]
//  [64,...)  float  xl[B][S_][LDC]   (~917 KB)

__global__ void k_init(uint32_t* tbits, float* wsum, uint32_t* wcnt) {
  int t = threadIdx.x;
  if (t < B_) { tbits[t] = 0x7F800000u; wsum[t] = 0.f; wcnt[t] = 0u; }
}

// One wave32 per row: exact K-th largest of the 512 row values via binary
// search on non-negative float bit patterns (order-preserving).
__global__ __launch_bounds__(256) void k_topk(const float* __restrict__ attn,
                                              uint32_t* __restrict__ tbits) {
  int wave = threadIdx.x >> 5, lane = threadIdx.x & 31;
  int row  = blockIdx.x * 8 + wave;             // 0 .. B*S-1
  int b = row >> 9, r = row & (S_ - 1);
  const float* wrow = attn + ((size_t)b * S_ + r) * S_;
  uint32_t bits[16];
#pragma unroll
  for (int i = 0; i < 16; ++i)
    bits[i] = __float_as_uint(wrow[lane + 32 * i]);   // coalesced

  uint32_t lo = 0u, hi = 0x3F800000u;   // uniform values live in [0,1)
  while (hi - lo > 1u) {
    uint32_t mid = (lo + hi) >> 1;
    int c = 0;
#pragma unroll
    for (int i = 0; i < 16; ++i) c += (bits[i] >= mid) ? 1 : 0;
#pragma unroll
    for (int off = 16; off > 0; off >>= 1) c += __shfl_xor(c, off, 32);
    if (c >= K_) lo = mid; else hi = mid;   // uniform branch
  }
  if (lane == 0) atomicMin(&tbits[b], lo);
}

// Masked sum / count over the full [S,S] matrix per batch.
__global__ __launch_bounds__(256) void k_stats(const float* __restrict__ attn,
                                               const uint32_t* __restrict__ tbits,
                                               float* __restrict__ wsum,
                                               uint32_t* __restrict__ wcnt) {
  int b = blockIdx.y;
  float thresh = __uint_as_float(tbits[b]);
  const float* w = attn + (size_t)b * S_ * S_;
  float s = 0.f; uint32_t c = 0;
  for (int idx = blockIdx.x * 256 + threadIdx.x; idx < S_ * S_;
       idx += gridDim.x * 256) {
    int r = idx >> 9, col = idx & (S_ - 1);
    float v = w[idx];
    bool m = (v > thresh) && (r != col);
    s += m ? v : 0.f;
    c += m ? 1u : 0u;
  }
#pragma unroll
  for (int off = 16; off > 0; off >>= 1) {
    s += __shfl_xor(s, off, 32);
    c += (uint32_t)__shfl_xor((int)c, off, 32);
  }
  __shared__ float    ss[8];
  __shared__ uint32_t sc[8];
  int wave = threadIdx.x >> 5, lane = threadIdx.x & 31;
  if (lane == 0) { ss[wave] = s; sc[wave] = c; }
  __syncthreads();
  if (threadIdx.x == 0) {
    float ts = 0.f; uint32_t tc = 0;
    for (int i = 0; i < 8; ++i) { ts += ss[i]; tc += sc[i]; }
    atomicAdd(&wsum[b], ts);
    atomicAdd(&wcnt[b], tc);
  }
}

// xl[b] = nodes[b] @ Wl + bl  via V_WMMA_F32_16X16X4_F32.
// Block = 8 waves; block owns (batch, n-tile, group of 8 m-tiles); the Wl
// column panel [768 x 16] is staged in LDS (48 KB) and shared by all waves.
// A layout  : lanes 0-15 -> M rows, VGPR0/1 = K {0,1} (lanes<16) / {2,3}
// B layout  : lanes 0-15 -> N cols, same K split
// C/D layout: VGPR i -> M = i (lanes<16) / 8+i, N = lane&15
__global__ __launch_bounds__(256) void k_gemm_xl(const float* __restrict__ value,
                                                 const float* __restrict__ Wl,
                                                 const float* __restrict__ bl,
                                                 float* __restrict__ xl) {
  __shared__ float Bs[H_ * 16];
  int blk = blockIdx.x;                 // 0 .. B*7*4-1
  int b = blk / 28; int rem = blk % 28;
  int ntile = rem >> 2, mgroup = rem & 3;
  int nbase = ntile * 16;
  int wave = threadIdx.x >> 5, lane = threadIdx.x & 31;

  for (int idx = threadIdx.x; idx < H_ * 16; idx += 256) {
    int kk = idx >> 4, cc = idx & 15;
    int col = nbase + cc;
    Bs[idx] = (col < C_) ? Wl[kk * C_ + col] : 0.f;
  }
  __syncthreads();

  int mtile = mgroup * 8 + wave;        // 0..31
  int r = lane & 15, khalf = lane >> 4; // r = A row / B col for this lane
  const float* A = value + ((size_t)b * S_ + mtile * 16 + r) * H_;

  v8f acc = {};
#pragma unroll 4
  for (int k = 0; k < H_; k += 4) {
    int ka = k + khalf * 2;
    v2f a;  a.x  = A[ka];              a.y  = A[ka + 1];
    v2f bm; bm.x = Bs[ka * 16 + r];    bm.y = Bs[(ka + 1) * 16 + r];
    acc = __builtin_amdgcn_wmma_f32_16x16x4_f32(
        /*neg_a=*/false, a, /*neg_b=*/false, bm,
        /*c_mod=*/(short)0, acc, /*reuse_a=*/false, /*reuse_b=*/false);
  }

  int col = nbase + r;
  float bias = (col < C_) ? bl[col] : 0.f;
  int mrow0 = mtile * 16 + khalf * 8;
  float* out = xl + (size_t)b * S_ * LDC;
#pragma unroll
  for (int i = 0; i < 8; ++i)
    out[(mrow0 + i) * LDC + col] = acc[i] + bias;   // padded cols hold 0+0
}

// Per-batch tail: xr[0], CLS-column GATv2 scoring, softmax over sources,
// message aggregation, ReLU, and the final C x 2 head.
__global__ __launch_bounds__(256) void k_finalize(
    const float* __restrict__ value, const float* __restrict__ attn,
    const float* __restrict__ Wr,    const float* __restrict__ br,
    const float* __restrict__ We,    const float* __restrict__ att,
    const float* __restrict__ bias_gat, const float* __restrict__ Wfc,
    const float* __restrict__ bfc,
    const float* __restrict__ xl,
    const uint32_t* __restrict__ tbits, const float* __restrict__ wsum,
    const uint32_t* __restrict__ wcnt,
    float* __restrict__ out) {
  int b = blockIdx.x, tid = threadIdx.x;
  int wave = tid >> 5, lane = tid & 31;
  __shared__ float s_xr0[C_];
  __shared__ float s_ea[S_];
  __shared__ float s_out[C_];
  __shared__ float s_red[8];

  float thresh = __uint_as_float(tbits[b]);
  uint32_t cnt = wcnt[b];
  float mean_w = wsum[b] / (float)(cnt > 0u ? cnt : 1u);

  // xr[0] = nodes[0] @ Wr + br  (only target column 0 is ever consumed)
  const float* n0 = value + (size_t)b * S_ * H_;
  for (int c = tid; c < C_; c += 256) {
    float acc = br[c];
    for (int h = 0; h < H_; ++h) acc += n0[h] * Wr[h * C_ + c];
    s_xr0[c] = acc;
  }
  __syncthreads();

  // e[j] for target 0, one wave per j (strided)
  const float* xlb = xl + (size_t)b * S_ * LDC;
  for (int j = wave; j < S_; j += 8) {
    float wj0 = attn[((size_t)b * S_ + j) * S_];         // w[j,0]
    float ew  = (j == 0) ? mean_w : wj0;                 // self-loop fill
    float part = 0.f;
    for (int c = lane; c < C_; c += 32) {
      float v = xlb[j * LDC + c] + s_xr0[c] + ew * We[c];
      v = (v > 0.f) ? v : 0.2f * v;                      // leaky_relu(0.2)
      part += v * att[c];
    }
#pragma unroll
    for (int off = 16; off > 0; off >>= 1) part += __shfl_xor(part, off, 32);
    bool keep = (j == 0) || (wj0 > thresh);
    if (lane == 0) s_ea[j] = keep ? part : NEGV;
  }
  __syncthreads();

  // softmax over sources j
  float m = NEGV;
  for (int j = tid; j < S_; j += 256) m = fmaxf(m, s_ea[j]);
#pragma unroll
  for (int off = 16; off > 0; off >>= 1) m = fmaxf(m, __shfl_xor(m, off, 32));
  if (lane == 0) s_red[wave] = m;
  __syncthreads();
  m = s_red[0];
  for (int i = 1; i < 8; ++i) m = fmaxf(m, s_red[i]);

  float sum = 0.f;
  for (int j = tid; j < S_; j += 256) sum += __expf(s_ea[j] - m);
#pragma unroll
  for (int off = 16; off > 0; off >>= 1) sum += __shfl_xor(sum, off, 32);
  __syncthreads();
  if (lane == 0) s_red[wave] = sum;
  __syncthreads();
  sum = 0.f;
  for (int i = 0; i < 8; ++i) sum += s_red[i];
  float inv = 1.f / sum;
  for (int j = tid; j < S_; j += 256) s_ea[j] = __expf(s_ea[j] - m) * inv;
  __syncthreads();

  // out[ch] = relu(sum_j alpha[j] * xl[j,ch] + bias_gat[ch])
  for (int ch = tid; ch < C_; ch += 256) {
    float acc = bias_gat[ch];
    for (int j = 0; j < S_; ++j) acc += s_ea[j] * xlb[j * LDC + ch];
    s_out[ch] = fmaxf(acc, 0.f);
  }
  __syncthreads();

  if (tid < 2) {
    float acc = bfc[tid];
    for (int ch = 0; ch < C_; ++ch) acc += s_out[ch] * Wfc[ch * 2 + tid];
    out[b * 2 + tid] = acc;
  }
}

extern "C" void kernel_launch(void* const* d_in, const int* in_sizes, int n_in,
                              void* d_out, int out_size, void* d_ws, size_t ws_size,
                              hipStream_t stream) {
  const float* value    = (const float*)d_in[0];
  const float* attn     = (const float*)d_in[1];
  const float* Wl       = (const float*)d_in[2];
  const float* bl       = (const float*)d_in[3];
  const float* Wr       = (const float*)d_in[4];
  const float* br       = (const float*)d_in[5];
  const float* We       = (const float*)d_in[6];
  const float* att      = (const float*)d_in[7];
  const float* bias_gat = (const float*)d_in[8];
  const float* Wfc      = (const float*)d_in[9];
  const float* bfc      = (const float*)d_in[10];
  float* out = (float*)d_out;

  char* ws = (char*)d_ws;
  uint32_t* tbits = (uint32_t*)(ws + 0);
  float*    wsum  = (float*)(ws + 16);
  uint32_t* wcnt  = (uint32_t*)(ws + 32);
  float*    xl    = (float*)(ws + 64);

  hipLaunchKernelGGL(k_init, dim3(1), dim3(32), 0, stream, tbits, wsum, wcnt);
  hipLaunchKernelGGL(k_topk, dim3(B_ * S_ / 8), dim3(256), 0, stream, attn, tbits);
  hipLaunchKernelGGL(k_stats, dim3(32, B_), dim3(256), 0, stream,
                     attn, tbits, wsum, wcnt);
  hipLaunchKernelGGL(k_gemm_xl, dim3(B_ * 7 * 4), dim3(256), 0, stream,
                     value, Wl, bl, xl);
  hipLaunchKernelGGL(k_finalize, dim3(B_), dim3(256), 0, stream,
                     value, attn, Wr, br, We, att, bias_gat, Wfc, bfc,
                     xl, tbits, wsum, wcnt, out);
}